// HDSymbolicAttention_43087111913928
// MI455X (gfx1250) — compile-verified
//
#include <hip/hip_runtime.h>
#include <hip/hip_bf16.h>
#include <math.h>

typedef float v2f __attribute__((ext_vector_type(2)));
typedef float v4f __attribute__((ext_vector_type(4)));
typedef float v8f __attribute__((ext_vector_type(8)));

#define D 1024
#define KIN 64
#define KLEN 961
#define PADC 960
#define BT 16              // batches per workgroup (WMMA M)
#define VPSTRIDE (D + 8)   // LDS row padding to stagger banks
#define NFRAG (2 * 64 * 32 * 16)   // v2f elements in fragment-packed W

__device__ __forceinline__ float silu(float x) {
    return x / (1.0f + __expf(-x));
}

// ---------------------------------------------------------------------------
// Kernel 1: expand h directly into WMMA-fragment-major W table and compute
// sp (2 x 1024).  Fragment layout (v2f elements):
//   Wf[((c*64 + ct)*32 + lane)*16 + s] = { W[c][kb][col], W[c][kb+1][col] }
//   kb = 4*s + (lane>=16 ? 2 : 0), col = ct*16 + (lane&15)
//   W[c][j][t] = h[c, j+960-t] if 0 <= j+960-t < 961 else 0
// Each lane's 16 K-step fragments occupy 128 contiguous bytes -> b128 loads.
// ---------------------------------------------------------------------------
__global__ __launch_bounds__(256) void hdsym_prep(
    const float* __restrict__ h,       // (2, 961)
    const float* __restrict__ symbols, // (2, 64)
    const float* __restrict__ gamma,   // (1024,)
    const float* __restrict__ beta,    // (1024,)
    v2f* __restrict__ Wf,              // NFRAG v2f = 512 KB
    float* __restrict__ sp)            // (2, 1024)
{
    const int tid = threadIdx.x;

    for (int idx = tid; idx < NFRAG; idx += 256) {
        int s  = idx & 15;
        int l  = (idx >> 4) & 31;
        int ct = (idx >> 9) & 63;
        int c  = idx >> 15;
        int col = ct * 16 + (l & 15);
        int kb  = 4 * s + ((l >> 4) ? 2 : 0);
        int k0  = kb + PADC - col;
        v2f w;
        w.x = (k0 >= 0 && k0 < KLEN)     ? h[c * KLEN + k0]     : 0.0f;
        w.y = (k0 + 1 >= 0 && k0 + 1 < KLEN) ? h[c * KLEN + k0 + 1] : 0.0f;
        Wf[idx] = w;
    }

    __shared__ float ys[2][D];
    __shared__ float red[2][2]; // {mu, rstd} per channel

    // ysym[a][t] = sum_j symbols[a,j] * h[a, j+960-t]
    for (int idx = tid; idx < 2 * D; idx += 256) {
        int t = idx & (D - 1);
        int a = idx >> 10;
        int j0 = (t - PADC) > 0 ? (t - PADC) : 0;
        int j1 = t < (KIN - 1) ? t : (KIN - 1);
        float acc = 0.0f;
        for (int j = j0; j <= j1; ++j)
            acc += symbols[a * KIN + j] * h[a * KLEN + (j + PADC - t)];
        ys[a][t] = acc;
    }
    __syncthreads();

    const int wave = tid >> 5;
    const int lane = tid & 31;
    if (wave < 2) {
        float s = 0.0f, ss = 0.0f;
        for (int t = lane; t < D; t += 32) {
            float v = ys[wave][t];
            s += v; ss += v * v;
        }
        #pragma unroll
        for (int m = 16; m >= 1; m >>= 1) {
            s  += __shfl_xor(s,  m, 32);
            ss += __shfl_xor(ss, m, 32);
        }
        float mu = s * (1.0f / D);
        float var = ss * (1.0f / D) - mu * mu;
        if (lane == 0) { red[wave][0] = mu; red[wave][1] = rsqrtf(var + 1e-3f); }
    }
    __syncthreads();

    for (int idx = tid; idx < 2 * D; idx += 256) {
        int t = idx & (D - 1);
        int a = idx >> 10;
        float ln = gamma[t] * (ys[a][t] - red[a][0]) * red[a][1] + beta[t];
        sp[idx] = silu(ln);
    }
}

// ---------------------------------------------------------------------------
// Kernel 2: per 16-batch tile: WMMA matmul (f32 16x16x4), LN+SiLU, 2x2 sign
// attention, SiLU output.  256 threads = 8 wave32s.
// ---------------------------------------------------------------------------
__global__ __launch_bounds__(256) void hdsym_main(
    const float* __restrict__ values,  // (B, 2, 64)
    const float* __restrict__ gamma,
    const float* __restrict__ beta,
    const v2f* __restrict__ Wf,        // fragment-packed W
    const float* __restrict__ sp_g,    // (2, 1024)
    float* __restrict__ out)           // (B, 2, 1024)
{
    extern __shared__ float lds[];
    float* vp = lds;                        // 32 rows x VPSTRIDE (y -> vp in place)
    float* xs = vp + 32 * VPSTRIDE;         // 16 x 2 x 64
    float* sp = xs + BT * 2 * KIN;          // 2 x 1024
    float* Sm = sp + 2 * D;                 // 64: S[bt][a][c]
    float* Sc = Sm + 64;                    // 64: softmax scores

    const int tid  = threadIdx.x;
    const int wave = tid >> 5;
    const int lane = tid & 31;
    const int l15  = lane & 15;
    const int hi   = lane >> 4;             // lanes 16..31
    const long b0  = (long)blockIdx.x * BT;

    // stage X tile (streaming: non-temporal) and sp (L2-hot: normal)
    for (int i = tid; i < BT * 2 * KIN; i += 256)
        xs[i] = __builtin_nontemporal_load(&values[b0 * 2 * KIN + i]);
    for (int i = tid; i < 2 * D; i += 256)
        sp[i] = sp_g[i];
    __syncthreads();

    // ---- WMMA: wave w -> channel w>>2, 16 column tiles ----
    const int ch = wave >> 2;

    // A fragments: 16x4 f32 layout -> lanes 0-15 hold M=lane {K0,K1},
    // lanes 16-31 hold M=lane-16 {K2,K3}.  Preload all 16 K-steps.
    v2f afrag[16];
    #pragma unroll
    for (int s = 0; s < 16; ++s) {
        int kb = 4 * s + (hi ? 2 : 0);
        const float* xr = xs + (l15 * 2 + ch) * KIN + kb;
        afrag[s].x = xr[0];
        afrag[s].y = xr[1];
    }

    const v2f* Wfc = Wf + (long)ch * 64 * 32 * 16;
    for (int i = 0; i < 16; ++i) {
        const int ct = (wave & 3) * 16 + i;
        const int colbase = ct << 4;
        // this lane's 16 B fragments: 128 contiguous bytes -> 8x b128
        const v4f* bp = (const v4f*)(Wfc + ((long)ct * 32 + lane) * 16);
        v8f acc = {};
        #pragma unroll
        for (int j = 0; j < 8; ++j) {
            v4f q = bp[j];
            v2f bf0; bf0.x = q.x; bf0.y = q.y;
            v2f bf1; bf1.x = q.z; bf1.y = q.w;
            acc = __builtin_amdgcn_wmma_f32_16x16x4_f32(
                false, afrag[2 * j],     false, bf0, (short)0, acc, false, false);
            acc = __builtin_amdgcn_wmma_f32_16x16x4_f32(
                false, afrag[2 * j + 1], false, bf1, (short)0, acc, false, false);
        }
        // C/D layout: lane = column (l15), VGPR r -> row r (lo) / r+8 (hi)
        #pragma unroll
        for (int r = 0; r < 8; ++r) {
            int m = hi ? (r + 8) : r;                     // batch-local row
            vp[(m * 2 + ch) * VPSTRIDE + colbase + l15] = acc[r];
        }
    }
    __syncthreads();

    // ---- LayerNorm + SiLU per row (32 rows, 4 per wave) ----
    for (int rr = wave * 4; rr < wave * 4 + 4; ++rr) {
        float* yr = vp + rr * VPSTRIDE;
        float s = 0.0f, ss = 0.0f;
        for (int t = lane; t < D; t += 32) {
            float v = yr[t];
            s += v; ss += v * v;
        }
        #pragma unroll
        for (int m = 16; m >= 1; m >>= 1) {
            s  += __shfl_xor(s,  m, 32);
            ss += __shfl_xor(ss, m, 32);
        }
        float mu   = s * (1.0f / D);
        float rstd = rsqrtf(ss * (1.0f / D) - mu * mu + 1e-3f);
        for (int t = lane; t < D; t += 32) {
            float ln = gamma[t] * (yr[t] - mu) * rstd + beta[t];
            yr[t] = silu(ln);
        }
    }
    __syncthreads();

    // ---- S[bt][a][c] = mean_d sign(vp[c,d]) * sign(vp[c,d] + sp[a,d]) ----
    for (int q = wave * 8; q < wave * 8 + 8; ++q) {
        int bt = q >> 2, a = (q >> 1) & 1, cc = q & 1;
        const float* vr = vp + (bt * 2 + cc) * VPSTRIDE;
        const float* sr = sp + a * D;
        float s = 0.0f;
        for (int t = lane; t < D; t += 32) {
            float x = vr[t];
            float y = x + sr[t];
            float sx = (float)((x > 0.0f) - (x < 0.0f));
            float sy = (float)((y > 0.0f) - (y < 0.0f));
            s += sx * sy;
        }
        #pragma unroll
        for (int m = 16; m >= 1; m >>= 1) s += __shfl_xor(s, m, 32);
        if (lane == 0) Sm[q] = s * (1.0f / D);
    }
    __syncthreads();

    // ---- softmax over c (2-wide), per (bt, a) ----
    if (tid < 32) {
        int base = tid * 2;   // tid = bt*2 + a
        float s0 = Sm[base], s1 = Sm[base + 1];
        float mx = fmaxf(s0, s1);
        float e0 = __expf(s0 - mx), e1 = __expf(s1 - mx);
        float inv = 1.0f / (e0 + e1);
        Sc[base] = e0 * inv; Sc[base + 1] = e1 * inv;
    }
    __syncthreads();

    // ---- out[b, a, d] = silu((sc0*vp[0,d] + sc1*vp[1,d]) * sp[a,d]) ----
    // streaming store: keep L2 for Wf/sp
    for (int idx = tid; idx < BT * 2 * D; idx += 256) {
        int d  = idx & (D - 1);
        int a  = (idx >> 10) & 1;
        int bt = idx >> 11;
        float sc0 = Sc[(bt * 2 + a) * 2 + 0];
        float sc1 = Sc[(bt * 2 + a) * 2 + 1];
        float attn = sc0 * vp[(bt * 2 + 0) * VPSTRIDE + d]
                   + sc1 * vp[(bt * 2 + 1) * VPSTRIDE + d];
        float v = attn * sp[a * D + d];
        __builtin_nontemporal_store(silu(v), &out[(b0 + bt) * 2 * D + a * D + d]);
    }
}

// ---------------------------------------------------------------------------
extern "C" void kernel_launch(void* const* d_in, const int* in_sizes, int n_in,
                              void* d_out, int out_size, void* d_ws, size_t ws_size,
                              hipStream_t stream) {
    const float* values  = (const float*)d_in[0]; // (16384, 2, 64)
    const float* h       = (const float*)d_in[1]; // (2, 961)
    const float* gamma   = (const float*)d_in[2]; // (1024,)
    const float* beta    = (const float*)d_in[3]; // (1024,)
    const float* symbols = (const float*)d_in[4]; // (2, 64)
    float* out = (float*)d_out;

    v2f*   Wf = (v2f*)d_ws;                 // NFRAG v2f = 512 KB
    float* sp = (float*)(Wf + NFRAG);       // 2*1024 f32

    hipLaunchKernelGGL(hdsym_prep, dim3(1), dim3(256), 0, stream,
                       h, symbols, gamma, beta, Wf, sp);

    const int B = in_sizes[0] / (2 * KIN);   // 16384
    const size_t shmem = (size_t)(32 * VPSTRIDE + BT * 2 * KIN + 2 * D + 128) * sizeof(float);
    hipLaunchKernelGGL(hdsym_main, dim3(B / BT), dim3(256), shmem, stream,
                       values, gamma, beta, Wf, sp, out);
}